// SKA_40106404610132
// MI455X (gfx1250) — compile-verified
//
#include <hip/hip_runtime.h>
#include <stdint.h>

// Problem constants (match reference setup_inputs)
#define BB   8
#define CC   512
#define LL   4096
#define WCH  64
#define KSZ  7
#define GRP  8          // CC / WCH
#define PADL 3          // (KSZ-1)/2
#define TL   512        // l-positions per block tile
#define NTHR 256        // 8 wave32s per block

typedef float v4f __attribute__((ext_vector_type(4)));

__global__ __launch_bounds__(NTHR) void
ska_dynconv_kernel(const float* __restrict__ x,
                   const float* __restrict__ w,
                   float* __restrict__ out) {
    // LDS tile of weights: w[b, wc, 0..6, l0 : l0+TL]  (7*512*4 = 14336 B)
    __shared__ float sw[KSZ * TL];

    const int tid    = threadIdx.x;
    const int ntiles = LL / TL;                    // 8
    const int blk    = blockIdx.x;                 // b*(WCH*ntiles) + wc*ntiles + tile
    const int tile   = blk % ntiles;
    const int wc     = (blk / ntiles) % WCH;
    const int b      = blk / (ntiles * WCH);
    const int l0     = tile * TL;

    // ---------------------------------------------------------------
    // Stage w tile into LDS via CDNA5 async global->LDS DMA (ASYNCcnt)
    // ---------------------------------------------------------------
    const float* wsrc = w + ((size_t)b * WCH + wc) * (size_t)(KSZ * LL);
    const uint64_t wbase64 = (uint64_t)(uintptr_t)wsrc;
    // Flat shared-memory address: bits [31:0] are the raw LDS byte offset
    // (LDS aperture base lives in bits [63:32], ISA sec 10.2).
    const uint32_t swbase = (uint32_t)(uintptr_t)&sw[0];

    // KSZ*TL/4 = 896 float4 elements to stage
    for (int i = tid; i < (KSZ * TL) / 4; i += NTHR) {
        const int k = i / (TL / 4);
        const int j = i % (TL / 4);
        const uint32_t goff = (uint32_t)((k * LL + l0 + j * 4) * sizeof(float));
        const uint32_t loff = swbase + (uint32_t)((k * TL + j * 4) * sizeof(float));
        asm volatile("global_load_async_to_lds_b128 %0, %1, %2 offset:0"
                     :
                     : "v"(loff), "v"(goff), "s"(wbase64)
                     : "memory");
    }
    asm volatile("s_wait_asynccnt 0" ::: "memory");
    __syncthreads();

    // ---------------------------------------------------------------
    // Compute: wave `wave` handles channel ch = wc*8 + wave (c/8 == wc)
    // Each lane: 4 chunks of 4 outputs -> TL outputs per wave.
    // ---------------------------------------------------------------
    const int wave = tid >> 5;
    const int lane = tid & 31;
    const int ch   = wc * GRP + wave;

    const float* xrow = x   + ((size_t)b * CC + ch) * (size_t)LL;
    float*       orow = out + ((size_t)b * CC + ch) * (size_t)LL;

#pragma unroll
    for (int j = 0; j < TL / (32 * 4); ++j) {
        const int lidx = (j * 32 + lane) * 4;   // position within tile
        const int lb   = l0 + lidx;             // first of 4 output l's

        // x[lb-4 .. lb+7] -> xv[0..11]; output t uses xv[t + k + 1]
        float xv[12];
        if (lb >= 4 && lb + 8 <= LL) {
            const v4f a = *(const v4f*)(xrow + lb - 4);
            const v4f m = *(const v4f*)(xrow + lb);
            const v4f c = *(const v4f*)(xrow + lb + 4);
            xv[0] = a.x;  xv[1] = a.y;  xv[2]  = a.z;  xv[3]  = a.w;
            xv[4] = m.x;  xv[5] = m.y;  xv[6]  = m.z;  xv[7]  = m.w;
            xv[8] = c.x;  xv[9] = c.y;  xv[10] = c.z;  xv[11] = c.w;
        } else {
#pragma unroll
            for (int t = 0; t < 12; ++t) {
                const int xi = lb - 4 + t;
                xv[t] = (xi >= 0 && xi < LL) ? xrow[xi] : 0.0f;
            }
        }

        float a0 = 0.f, a1 = 0.f, a2 = 0.f, a3 = 0.f;
#pragma unroll
        for (int k = 0; k < KSZ; ++k) {
            const v4f wv = *(const v4f*)(&sw[k * TL + lidx]);
            a0 = __builtin_fmaf(xv[k + 1], wv.x, a0);
            a1 = __builtin_fmaf(xv[k + 2], wv.y, a1);
            a2 = __builtin_fmaf(xv[k + 3], wv.z, a2);
            a3 = __builtin_fmaf(xv[k + 4], wv.w, a3);
        }

        // Streaming (write-once) output: non-temporal store, keep L2 clean.
        v4f o;
        o.x = a0; o.y = a1; o.z = a2; o.w = a3;
        __builtin_nontemporal_store(o, (v4f*)(orow + lb));
    }
}

extern "C" void kernel_launch(void* const* d_in, const int* in_sizes, int n_in,
                              void* d_out, int out_size, void* d_ws, size_t ws_size,
                              hipStream_t stream) {
    (void)in_sizes; (void)n_in; (void)d_ws; (void)ws_size; (void)out_size;
    const float* x = (const float*)d_in[0];   // [B, C, L]
    const float* w = (const float*)d_in[1];   // [B, WC, KS, L]
    float* out = (float*)d_out;               // [B, C, L]

    const int ntiles = LL / TL;               // 8
    dim3 grid(BB * WCH * ntiles);             // 4096 blocks
    dim3 block(NTHR);                         // 8 wave32s
    ska_dynconv_kernel<<<grid, block, 0, stream>>>(x, w, out);
}